// DSMoE_33028298506248
// MI455X (gfx1250) — compile-verified
//
#include <hip/hip_runtime.h>
#include <hip/hip_bf16.h>
#include <math.h>
#include <stdint.h>

typedef __attribute__((ext_vector_type(2))) float v2f;
typedef __attribute__((ext_vector_type(8))) float v8f;

#define C_DIM 256
#define H_DIM 1024
#define N_EXPERTS 32
#define TILE_M 32
#define MAX_TOK 4096
// padded LDS row strides (floats) to avoid bank conflicts on column reads
#define XS_STRIDE 260   // 260 % 64 = 4 -> 16 rows hit distinct banks; 16B aligned rows
#define HS_STRIDE 1028  // 1028 % 64 = 4

// ---- CDNA5 async global->LDS copy (ASYNCcnt-tracked), per CDNA5 ISA ----
__device__ __forceinline__ void async_load_b128(uint32_t lds_off, const void* gptr) {
    asm volatile("global_load_async_to_lds_b128 %0, %1, off"
                 :: "v"(lds_off), "v"(gptr) : "memory");
}
__device__ __forceinline__ void wait_async0() {
    asm volatile("s_wait_asynccnt 0x0" ::: "memory");
}

// ---------------- kernel 0: zero the per-expert counters ----------------
__global__ void moe_zero_counts(int* counts) {
    if (threadIdx.x < N_EXPERTS) counts[threadIdx.x] = 0;
}

// ---------------- kernel 1: router (one wave32 per token) ----------------
__global__ void moe_router(const float* __restrict__ x,
                           const float* __restrict__ gate_W,
                           const float* __restrict__ gate_b,
                           const float* __restrict__ expert_bias,
                           float* __restrict__ out,       // [N, C] (zeroed here)
                           float* __restrict__ rw,        // [N, 32] router weights
                           int* __restrict__ counts,
                           int* __restrict__ tokens,
                           float* __restrict__ wts,
                           int N) {
    const int n = blockIdx.x;
    const int lane = threadIdx.x;   // 0..31, wave32

    // --- gate logits: 31 routed experts, lane 31 = -inf sentinel ---
    float logit = -INFINITY;
    if (lane < N_EXPERTS - 1) {
        float s = gate_b[lane];
        const float* xr = x + (size_t)n * C_DIM;
        #pragma unroll 4
        for (int c = 0; c < C_DIM; ++c)
            s += xr[c] * gate_W[c * (N_EXPERTS - 1) + lane];
        logit = s;
    }

    // --- softmax over 31 lanes (wave butterfly reductions) ---
    float m = logit;
    for (int off = 16; off > 0; off >>= 1)
        m = fmaxf(m, __shfl_xor(m, off, 32));
    float ex = (lane < N_EXPERTS - 1) ? expf(logit - m) : 0.0f;
    float ssum = ex;
    for (int off = 16; off > 0; off >>= 1)
        ssum += __shfl_xor(ssum, off, 32);
    float gate = ex / ssum;

    float biased = (lane < N_EXPERTS - 1) ? (gate + expert_bias[lane]) : -INFINITY;

    // --- top-3 (tie -> lower index, matching lax.top_k) ---
    float cur = biased;
    int   sel_i[3];
    float sel_v[3];
    #pragma unroll
    for (int j = 0; j < 3; ++j) {
        float bv = cur;
        int   bi = lane;
        for (int off = 16; off > 0; off >>= 1) {
            float ov = __shfl_xor(bv, off, 32);
            int   oi = __shfl_xor(bi, off, 32);
            if (ov > bv || (ov == bv && oi < bi)) { bv = ov; bi = oi; }
        }
        sel_i[j] = bi;
        sel_v[j] = bv;
        if (lane == bi) cur = -INFINITY;
    }
    const float vsum = sel_v[0] + sel_v[1] + sel_v[2];

    // --- dense router-weight row (output #2): expert id == lane ---
    float w = 0.0f;
    if (lane == 0) w = 0.25f;                       // shared expert
    #pragma unroll
    for (int j = 0; j < 3; ++j)
        if (lane == sel_i[j] + 1) w += (sel_v[j] / vsum) * 0.75f;
    rw[(size_t)n * N_EXPERTS + lane] = w;

    // --- zero this token's output row (stage 2 accumulates atomically) ---
    float* orow = out + (size_t)n * C_DIM;
    #pragma unroll
    for (int i = lane; i < C_DIM; i += 32) orow[i] = 0.0f;

    // --- append (token, weight) to the 4 selected experts' lists ---
    if (lane < 4) {
        int   e, slot;
        float wv;
        if (lane == 0) { e = 0;               wv = 0.25f; }
        else           { e = sel_i[lane - 1] + 1;
                         wv = (sel_v[lane - 1] / vsum) * 0.75f; }
        slot = atomicAdd(&counts[e], 1);
        tokens[e * MAX_TOK + slot] = n;
        wts[e * MAX_TOK + slot]    = wv;
    }
}

// ---------------- kernel 2: grouped expert FFN via f32 WMMA ----------------
// grid: (ceil(N/32) tiles, 32 experts), 256 threads = 8 waves
__global__ __launch_bounds__(256)
void moe_expert_ffn(const float* __restrict__ x,
                    const float* __restrict__ W1,   // [E, C, H] row-major
                    const float* __restrict__ b1,   // [E, H]
                    const float* __restrict__ W2,   // [E, H, C] row-major
                    const float* __restrict__ b2,   // [E, C]
                    float* __restrict__ out,        // [N, C], atomic accum
                    const int* __restrict__ counts,
                    const int* __restrict__ tokens,
                    const float* __restrict__ wts) {
    __shared__ float xs[TILE_M * XS_STRIDE];   // gathered x tile  (~33 KB)
    __shared__ float hs[TILE_M * HS_STRIDE];   // gelu(hid) tile   (~132 KB)
    __shared__ int   tok_s[TILE_M];
    __shared__ float wt_s[TILE_M];

    const int e    = blockIdx.y;
    const int cnt  = counts[e];
    const int m0   = blockIdx.x * TILE_M;
    if (m0 >= cnt) return;                     // uniform: EXEC stays all-1s

    const int tid  = threadIdx.x;
    const int wave = tid >> 5;
    const int lane = tid & 31;
    const int lrow = lane & 15;                // A-matrix M index / B column
    const int koff = (lane < 16) ? 0 : 2;      // K sub-offset per 16x16x4 layout
    const int mofs = (lane < 16) ? 0 : 8;      // C/D layout row offset

    // --- fetch token ids & gate weights for this tile ---
    if (tid < TILE_M) {
        int gi = m0 + tid;
        if (gi < cnt) { tok_s[tid] = tokens[e * MAX_TOK + gi];
                        wt_s[tid]  = wts[e * MAX_TOK + gi]; }
        else          { tok_s[tid] = -1; wt_s[tid] = 0.0f; }
    }
    __syncthreads();

    // --- gather 32 x-rows into LDS via async global->LDS b128 copies ---
    // padding rows reuse token 0's row: finite garbage, WMMA rows independent,
    // masked at the epilogue (tk<0, wt=0) -> all lanes issue unconditionally.
    {
        int r   = tid >> 3;                    // row 0..31
        int cb  = (tid & 7) * 32;              // 32 floats (8 x b128) per thread
        int tk  = tok_s[r];
        int tks = (tk < 0) ? 0 : tk;
        const float* src = x + (size_t)tks * C_DIM + cb;
        uint32_t dst = (uint32_t)(uintptr_t)&xs[r * XS_STRIDE + cb];
        #pragma unroll
        for (int i = 0; i < 8; ++i)
            async_load_b128(dst + i * 16, src + i * 4);
        wait_async0();
    }
    __syncthreads();

    // =================== GEMM1: hid = gelu(x @ W1[e] + b1[e]) ===================
    // wave owns a 32 x 128 slice of hid: 2 M-subtiles x 8 N-tiles.
    // Each B fragment feeds 2 WMMAs -> halves weight traffic per token.
    {
        const int nbase = wave * 128;
        v8f acc[2][8];
        #pragma unroll
        for (int s = 0; s < 2; ++s)
            #pragma unroll
            for (int t = 0; t < 8; ++t) acc[s][t] = (v8f)0.0f;

        const float* W1e = W1 + (size_t)e * C_DIM * H_DIM;
        for (int k4 = 0; k4 < C_DIM; k4 += 4) {
            v2f a0, a1;
            const float* ap0 = &xs[lrow * XS_STRIDE + k4 + koff];
            const float* ap1 = &xs[(16 + lrow) * XS_STRIDE + k4 + koff];
            a0.x = ap0[0]; a0.y = ap0[1];
            a1.x = ap1[0]; a1.y = ap1[1];
            const float* Bb = W1e + (size_t)(k4 + koff) * H_DIM + nbase + lrow;
            #pragma unroll
            for (int t = 0; t < 8; ++t) {
                v2f b;
                b.x = Bb[t * 16];
                b.y = Bb[t * 16 + H_DIM];
                acc[0][t] = __builtin_amdgcn_wmma_f32_16x16x4_f32(
                    false, a0, false, b, (short)0, acc[0][t], false, false);
                acc[1][t] = __builtin_amdgcn_wmma_f32_16x16x4_f32(
                    false, a1, false, b, (short)0, acc[1][t], false, false);
            }
        }

        // bias + tanh-GELU, write to LDS (C/D layout: vgpr r -> M=r(+8))
        #pragma unroll
        for (int s = 0; s < 2; ++s) {
            #pragma unroll
            for (int t = 0; t < 8; ++t) {
                int ncol = nbase + t * 16 + lrow;
                float bias = b1[e * H_DIM + ncol];
                #pragma unroll
                for (int r = 0; r < 8; ++r) {
                    float v = acc[s][t][r] + bias;
                    float g = 0.5f * v *
                              (1.0f + tanhf(0.7978845608028654f *
                                            (v + 0.044715f * v * v * v)));
                    hs[(s * 16 + r + mofs) * HS_STRIDE + ncol] = g;
                }
            }
        }
    }
    __syncthreads();

    // ============ GEMM2: out += wt * (hid @ W2[e] + b2[e]) (atomic) ============
    // wave owns 32 x 32 of the C=256 output: 2 M-subtiles x 2 N-tiles.
    {
        v8f acc2[2][2];
        #pragma unroll
        for (int s = 0; s < 2; ++s)
            #pragma unroll
            for (int t = 0; t < 2; ++t) acc2[s][t] = (v8f)0.0f;

        const float* W2e = W2 + (size_t)e * H_DIM * C_DIM;
        for (int k4 = 0; k4 < H_DIM; k4 += 4) {
            v2f a0, a1;
            const float* ap0 = &hs[lrow * HS_STRIDE + k4 + koff];
            const float* ap1 = &hs[(16 + lrow) * HS_STRIDE + k4 + koff];
            a0.x = ap0[0]; a0.y = ap0[1];
            a1.x = ap1[0]; a1.y = ap1[1];
            const float* Bb = W2e + (size_t)(k4 + koff) * C_DIM + lrow;
            #pragma unroll
            for (int t = 0; t < 2; ++t) {
                int nb = (wave * 2 + t) * 16;
                v2f b;
                b.x = Bb[nb];
                b.y = Bb[nb + C_DIM];
                acc2[0][t] = __builtin_amdgcn_wmma_f32_16x16x4_f32(
                    false, a0, false, b, (short)0, acc2[0][t], false, false);
                acc2[1][t] = __builtin_amdgcn_wmma_f32_16x16x4_f32(
                    false, a1, false, b, (short)0, acc2[1][t], false, false);
            }
        }

        #pragma unroll
        for (int s = 0; s < 2; ++s) {
            #pragma unroll
            for (int t = 0; t < 2; ++t) {
                int ncol = (wave * 2 + t) * 16 + lrow;
                float bb = b2[e * C_DIM + ncol];
                #pragma unroll
                for (int r = 0; r < 8; ++r) {
                    int row = s * 16 + r + mofs;
                    int tk  = tok_s[row];
                    if (tk >= 0) {
                        float contrib = wt_s[row] * (acc2[s][t][r] + bb);
                        atomicAdd(&out[(size_t)tk * C_DIM + ncol], contrib);
                    }
                }
            }
        }
    }
}

// ------------------------------ launcher ------------------------------
extern "C" void kernel_launch(void* const* d_in, const int* in_sizes, int n_in,
                              void* d_out, int out_size, void* d_ws, size_t ws_size,
                              hipStream_t stream) {
    const float* x           = (const float*)d_in[0];
    const float* gate_W      = (const float*)d_in[1];
    const float* gate_b      = (const float*)d_in[2];
    const float* expert_bias = (const float*)d_in[3];
    const float* W1          = (const float*)d_in[4];
    const float* b1          = (const float*)d_in[5];
    const float* W2          = (const float*)d_in[6];
    const float* b2          = (const float*)d_in[7];

    const int N = in_sizes[0] / C_DIM;   // 4096 tokens

    float* out = (float*)d_out;                       // [N, C]
    float* rw  = out + (size_t)N * C_DIM;             // [N, 32]

    int*   counts = (int*)d_ws;
    int*   tokens = counts + N_EXPERTS;
    float* wvals  = (float*)(tokens + N_EXPERTS * MAX_TOK);

    moe_zero_counts<<<1, 32, 0, stream>>>(counts);

    moe_router<<<N, 32, 0, stream>>>(x, gate_W, gate_b, expert_bias,
                                     out, rw, counts, tokens, wvals, N);

    dim3 grid2((N + TILE_M - 1) / TILE_M, N_EXPERTS);
    moe_expert_ffn<<<grid2, 256, 0, stream>>>(x, W1, b1, W2, b2,
                                              out, counts, tokens, wvals);
}